// DepAttention_36584531427854
// MI455X (gfx1250) — compile-verified
//
#include <hip/hip_runtime.h>
#include <hip/hip_bf16.h>
#include <math.h>

#define B_DIM 8
#define N_DIM 256
#define D_DIM 128

typedef float v2f __attribute__((ext_vector_type(2)));
typedef float v8f __attribute__((ext_vector_type(8)));

// ---------------------------------------------------------------------------
// Kernel A: gram[b,i,j] = <val[b,i,:], val[b,j,:]>  (dense fp32 GEMM V*V^T)
// One wave per 16x16 output tile, V_WMMA_F32_16X16X4_F32 over K=128.
//
// A (16x4 f32) lane layout: lanes 0-15 hold M=lane, {K=k,k+1} in v[0],v[1];
// lanes 16-31 hold M=lane-16, {K=k+2,k+3}.  B (4x16) mirrors A with N=lane
// (B[k,n] = val[jt+n,k]), so both sides use the identical load pattern.
// C/D: VGPR r = row (it + r) for lanes 0-15, (it + 8 + r) for lanes 16-31,
// column jt + (lane&15).
// ---------------------------------------------------------------------------
__global__ __launch_bounds__(256) void dep_attn_gram_wmma(
    const float* __restrict__ val, float* __restrict__ gram)
{
    const int lane = threadIdx.x & 31;
    const int wave = (int)((blockIdx.x * blockDim.x + threadIdx.x) >> 5);
    const int tiles_per_b = (N_DIM / 16) * (N_DIM / 16);   // 256
    const int b  = wave / tiles_per_b;
    const int t  = wave % tiles_per_b;
    const int it = (t >> 4) << 4;    // i-tile base
    const int jt = (t & 15) << 4;    // j-tile base

    const int row  = lane & 15;
    const int koff = (lane >> 4) << 1;    // 0 or 2

    const float* __restrict__ arow = val + (size_t)(b * N_DIM + it + row) * D_DIM;
    const float* __restrict__ brow = val + (size_t)(b * N_DIM + jt + row) * D_DIM;

    v8f acc = {0.f, 0.f, 0.f, 0.f, 0.f, 0.f, 0.f, 0.f};
    #pragma unroll 8
    for (int k = 0; k < D_DIM; k += 4) {
        v2f a  = *(const v2f*)(arow + k + koff);   // 8B-aligned global_load_b64
        v2f bb = *(const v2f*)(brow + k + koff);
        acc = __builtin_amdgcn_wmma_f32_16x16x4_f32(
            /*neg_a=*/false, a, /*neg_b=*/false, bb,
            /*c_mod=*/(short)0, acc, /*reuse_a=*/false, /*reuse_b=*/false);
    }

    const int mbase = it + ((lane >> 4) << 3);   // +0 for lanes 0-15, +8 for 16-31
    const int col   = jt + (lane & 15);
    #pragma unroll
    for (int r = 0; r < 8; ++r)
        gram[(size_t)(b * N_DIM + mbase + r) * N_DIM + col] = acc[r];
}

// ---------------------------------------------------------------------------
// Kernel B: sparse dep-dot + masked exp-normalize. One wave32 per (b,i) row.
// Only the ~5% of j with adj[b,i,j] != 0 ever touch dep_embed, cutting the
// dominant HBM traffic from 256MB to ~29MB. Row sum is reduced in-wave (no
// atomics) and normalization is fused, so the final output is written once.
// ---------------------------------------------------------------------------
__global__ __launch_bounds__(256) void dep_attn_sparse_row(
    const float* __restrict__ dep, const float* __restrict__ adj,
    const float* __restrict__ gram, float* __restrict__ out)
{
    const int lane = threadIdx.x & 31;
    const int wave = (int)((blockIdx.x * blockDim.x + threadIdx.x) >> 5);
    const int b = wave >> 8;          // wave / N_DIM  (B*N = 2048 rows total)
    const int i = wave & (N_DIM - 1);

    const float scale = 0.088388347648318447f;   // 1/sqrt(128)

    const size_t rowBase = (size_t)(b * N_DIM + i) * N_DIM;
    const float* __restrict__ adjRow  = adj  + rowBase;
    const float* __restrict__ gramRow = gram + rowBase;

    float ev[8];
    #pragma unroll
    for (int c = 0; c < 8; ++c) ev[c] = 0.f;
    float rs = 0.f;

    const int d0 = lane * 4;                       // this lane's 4 dims of D
    const size_t depIbase = rowBase * D_DIM;       // dep[b,i,j,:] = +j*D

    for (int c = 0; c < 8; ++c) {
        const int   jl   = c * 32 + lane;
        const float adjv = adjRow[jl];             // coalesced
        const float gv   = gramRow[jl];            // coalesced
        unsigned int m = (unsigned int)__ballot(adjv != 0.f);
        while (m) {
            const int jsub = __builtin_ctz(m);
            m &= m - 1u;
            const int j = c * 32 + jsub;
            if (m) {  // prefetch next active (i,j)/(j,i) vector pair
                const int jn = c * 32 + __builtin_ctz(m);
                __builtin_prefetch(dep + depIbase + (size_t)jn * D_DIM + d0, 0, 3);
                __builtin_prefetch(dep + ((size_t)(b * N_DIM + jn) * N_DIM + i) * D_DIM + d0, 0, 3);
            }
            // <dep[b,i,j,:], dep[b,j,i,:]> : two coalesced 512B vectors
            const float4 x = *(const float4*)(dep + depIbase + (size_t)j * D_DIM + d0);
            const float4 y = *(const float4*)(dep + ((size_t)(b * N_DIM + j) * N_DIM + i) * D_DIM + d0);
            float p = fmaf(x.x, y.x, fmaf(x.y, y.y, fmaf(x.z, y.z, x.w * y.w)));
            #pragma unroll
            for (int off = 16; off > 0; off >>= 1)
                p += __shfl_xor(p, off, 32);       // butterfly: all lanes get sum
            if (lane == jsub) {                    // owner lane keeps the result
                const float e = __expf((gv + p) * scale) * adjv;
                ev[c] = e;
                rs += e;
            }
        }
    }

    #pragma unroll
    for (int off = 16; off > 0; off >>= 1)
        rs += __shfl_xor(rs, off, 32);
    const float inv = 1.0f / (rs + 1e-10f);

    float* __restrict__ orow = out + rowBase;
    #pragma unroll
    for (int c = 0; c < 8; ++c)
        orow[c * 32 + lane] = ev[c] * inv;         // coalesced stores
}

extern "C" void kernel_launch(void* const* d_in, const int* in_sizes, int n_in,
                              void* d_out, int out_size, void* d_ws, size_t ws_size,
                              hipStream_t stream) {
    const float* val = (const float*)d_in[0];   // [B,N,D]
    const float* dep = (const float*)d_in[1];   // [B,N,N,D]
    const float* adj = (const float*)d_in[2];   // [B,N,N]
    float* out  = (float*)d_out;                // [B,N,N]
    float* gram = (float*)d_ws;                 // B*N*N floats = 2 MB scratch

    // Kernel A: 8 batches * 16*16 tiles = 2048 waves -> 256 blocks of 8 waves.
    dep_attn_gram_wmma<<<dim3(256), dim3(256), 0, stream>>>(val, gram);
    // Kernel B: B*N = 2048 row-waves -> 256 blocks of 8 waves.
    dep_attn_sparse_row<<<dim3(256), dim3(256), 0, stream>>>(dep, adj, gram, out);
}